// VectorizedSparseAttention_13932873908465
// MI455X (gfx1250) — compile-verified
//
#include <hip/hip_runtime.h>
#include <stdint.h>

// ---------------------------------------------------------------------------
// Top-k row sparsification (keep top S/2 per row, else -inf), MI455X / gfx1250.
//
// Bandwidth-bound: read attn (1.07GB) once, write out (1.07GB) once, mask
// (67MB) stays L2-resident across its 16 H-reuses  => ~95us roofline @23.3TB/s.
// No matmul in the op => no WMMA; the CDNA5 features that matter are wave32,
// async global->LDS staging, LDS atomics, and cross-lane shuffles.
//
// One 256-thread (8 x wave32) workgroup per row of length S=2048:
//   1) async global->LDS copy of the 8KB attn row
//      (global_load_async_to_lds_b128 + s_wait_asynccnt)
//   2) add mask (L2), map float -> order-preserving u32 key; keys stay in
//      8 VGPRs per thread
//   3) exact 4-pass MSB-first radix select: 256-bin LDS histogram (ds_add),
//      then a barrier-free single-wave suffix scan (8 bins/lane serial +
//      5-step __shfl_down across 32 lanes) -> exact 1024th-largest key
//   4) one vectorized b128 output pass: key > thr -> value, key == thr ->
//      admit up to remaining quota via LDS tie counter (ds_add_rtn), else -inf
// ---------------------------------------------------------------------------

namespace {
constexpr int kS       = 2048;            // row length (keys axis)
constexpr int kThreads = 256;             // 8 waves of 32
constexpr int kEPT     = kS / kThreads;   // 8 elements per thread
constexpr int kKeep    = kS / 2;          // top-k, k = 1024
}

__device__ __forceinline__ uint32_t f2key(float f) {
  uint32_t u = __float_as_uint(f);
  // order-preserving bijection float -> uint32 (descending float == descending uint)
  return u ^ ((u >> 31) ? 0xFFFFFFFFu : 0x80000000u);
}

__device__ __forceinline__ float key2f(uint32_t k) {
  return __uint_as_float(k ^ ((k >> 31) ? 0x80000000u : 0xFFFFFFFFu));
}

__global__ __launch_bounds__(kThreads) void VectorizedSparseAttention_topk_kernel(
    const float* __restrict__ aw, const float* __restrict__ mask,
    float* __restrict__ out, int hShift) {
  __shared__ __align__(16) float stage[kS];   // 8KB async-staged attn row
  __shared__ uint32_t hist[256];
  __shared__ uint32_t sBin;
  __shared__ uint32_t sK;
  __shared__ uint32_t sTie;

  const int tid = threadIdx.x;
  const int row = blockIdx.x;            // 0 .. B*H*S-1
  const int q   = row & (kS - 1);
  const int bh  = row >> 11;             // row / S
  const int b   = bh >> hShift;          // bh / H (H is a power of two)
  const size_t rowOff  = (size_t)row * kS;
  const size_t maskOff = ((size_t)b * kS + (size_t)q) * kS;

  // ---- 1) CDNA5 async global -> LDS copy of the row (two B128 per thread) --
  {
    uint32_t ldsBase = (uint32_t)(uintptr_t)(&stage[0]);  // low 32 bits = LDS offset
    uint32_t g0 = (uint32_t)(tid * 16);
    uint32_t g1 = g0 + (uint32_t)(kThreads * 16);
    uint32_t l0 = ldsBase + g0;
    uint32_t l1 = ldsBase + g1;
    const float* src = aw + rowOff;
    asm volatile(
        "global_load_async_to_lds_b128 %0, %2, %4\n\t"
        "global_load_async_to_lds_b128 %1, %3, %4\n\t"
        :
        : "v"(l0), "v"(l1), "v"(g0), "v"(g1), "s"(src)
        : "memory");
    asm volatile("s_wait_asynccnt 0" ::: "memory");
  }
  __syncthreads();

  // ---- 2) add mask, build sortable keys in registers -----------------------
  uint32_t key[kEPT];
  {
    const float* mrow = mask + maskOff;
    const int base = tid * kEPT;
#pragma unroll
    for (int j = 0; j < kEPT / 4; ++j) {
      float4 m = *(const float4*)(mrow + base + 4 * j);
      float4 a = *(const float4*)(&stage[base + 4 * j]);
      key[4 * j + 0] = f2key(a.x + m.x);
      key[4 * j + 1] = f2key(a.y + m.y);
      key[4 * j + 2] = f2key(a.z + m.z);
      key[4 * j + 3] = f2key(a.w + m.w);
    }
  }

  // ---- 3) exact MSB-first radix select for the kKeep-th largest key --------
  uint32_t prefix = 0;     // fixed high bits of the threshold key
  uint32_t kRem = kKeep;   // rank to locate within prefix-matching keys
#pragma unroll 1
  for (int pass = 0; pass < 4; ++pass) {
    const int shift = 24 - 8 * pass;
    const uint32_t pmask = (pass == 0) ? 0u : (0xFFFFFFFFu << (shift + 8));

    hist[tid] = 0u;
    __syncthreads();
#pragma unroll
    for (int j = 0; j < kEPT; ++j) {
      if ((key[j] & pmask) == prefix)
        atomicAdd(&hist[(key[j] >> shift) & 0xFFu], 1u);   // ds_add_u32
    }
    __syncthreads();

    // Single-wave (wave32) barrier-free 256-bin suffix scan:
    // lane l owns bins [8l, 8l+8); cross-lane suffix via 5 shfl_down steps.
    if (tid < 32) {
      const int lane = tid;
      uint32_t h[8];
      uint32_t tot = 0u;
#pragma unroll
      for (int i = 0; i < 8; ++i) {
        h[i] = hist[lane * 8 + i];
        tot += h[i];
      }
      // inclusive suffix sum across lanes: suf_l = sum_{l' >= l} tot_{l'}
      uint32_t suf = tot;
#pragma unroll
      for (int off = 1; off < 32; off <<= 1) {
        uint32_t other = __shfl_down(suf, off, 32);
        if (lane + off < 32) suf += other;
      }
      uint32_t above = __shfl_down(suf, 1, 32);   // sum over lanes > l
      if (lane == 31) above = 0u;
      // walk own bins from high to low: run = count of keys in bins > current
      uint32_t run = above;
#pragma unroll
      for (int i = 7; i >= 0; --i) {
        const uint32_t ge = run + h[i];
        if (run < kRem && kRem <= ge) {            // unique winner bin
          sBin = (uint32_t)(lane * 8 + i);
          sK   = kRem - run;
        }
        run += h[i];
      }
    }
    __syncthreads();
    prefix |= (sBin << shift);
    kRem = sK;
    // next pass's hist-zero + barrier separates this read from the next write
  }

  // ---- 4) emit: keep keys > thr, admit kRem of the == thr ties, else -inf --
  if (tid == 0) sTie = 0u;
  __syncthreads();
  const uint32_t thr   = prefix;
  const uint32_t quota = kRem;
  const uint32_t NEG_INF = 0xFF800000u;

  float* orow = out + rowOff;
  const int base = tid * kEPT;
#pragma unroll
  for (int j = 0; j < kEPT / 4; ++j) {
    float o[4];
#pragma unroll
    for (int c = 0; c < 4; ++c) {
      const uint32_t kk = key[4 * j + c];
      float v;
      if (kk > thr) {
        v = key2f(kk);
      } else if (kk == thr) {
        uint32_t slot = atomicAdd(&sTie, 1u);                // ds_add_rtn_u32
        v = (slot < quota) ? key2f(kk) : __uint_as_float(NEG_INF);
      } else {
        v = __uint_as_float(NEG_INF);
      }
      o[c] = v;
    }
    *(float4*)(orow + base + 4 * j) = make_float4(o[0], o[1], o[2], o[3]);
  }
}

extern "C" void kernel_launch(void* const* d_in, const int* in_sizes, int n_in,
                              void* d_out, int out_size, void* d_ws, size_t ws_size,
                              hipStream_t stream) {
  (void)n_in; (void)d_ws; (void)ws_size;
  const float* aw   = (const float*)d_in[0];   // [B,H,S,S] fp32
  const float* mask = (const float*)d_in[1];   // [B,1,S,S] fp32
  float* out = (float*)d_out;                  // [B,H,S,S] fp32

  const int rows = out_size / kS;              // B*H*S = 131072
  const int H    = in_sizes[0] / in_sizes[1];  // 16 (aw elems / mask elems)
  int hShift = 0;
  while ((1 << (hShift + 1)) <= H) ++hShift;   // log2(H), H is a power of two

  VectorizedSparseAttention_topk_kernel<<<rows, kThreads, 0, stream>>>(aw, mask, out, hShift);
}